// MPNN_63685775065690
// MI455X (gfx1250) — compile-verified
//
#include <hip/hip_runtime.h>

#define N_NODES 10000
#define N_EDGES 50000
#define EMBED   64
#define WCOLS   192          // [p | q | r] packed weight columns
#define NEG_SLOPE 0.01f

typedef float v2f __attribute__((ext_vector_type(2)));
typedef float v8f __attribute__((ext_vector_type(8)));

__device__ __forceinline__ float lrelu(float x) {
    // branchless leaky relu: x>0 -> x ; x<0 -> 0.01x
    return fmaxf(x, 0.0f) + NEG_SLOPE * fminf(x, 0.0f);
}

// ---------------------------------------------------------------- utilities
__global__ void zero_f32(float* __restrict__ p, int n) {
    int i = blockIdx.x * blockDim.x + threadIdx.x;
    if (i < n) p[i] = 0.0f;
}

// h0[n][j] = x[n] * node_w[j] + node_b[j]
__global__ void init_h(const float* __restrict__ x, const float* __restrict__ nw,
                       const float* __restrict__ nb, float* __restrict__ h) {
    int i = blockIdx.x * blockDim.x + threadIdx.x;
    if (i >= N_NODES * EMBED) return;
    int n = i >> 6, j = i & 63;
    h[i] = fmaf(x[n], nw[j], nb[j]);
}

// Pack W[64][192] = [ W1 | B1 | root ]  (row = input dim i, col = output dim o)
__global__ void build_w(const float* __restrict__ l1w, const float* __restrict__ l1b,
                        const float* __restrict__ root, float* __restrict__ W) {
    int i = blockIdx.x * blockDim.x + threadIdx.x;
    if (i >= EMBED * WCOLS) return;
    int r = i / WCOLS, c = i % WCOLS;
    float v;
    if (c < 64)       v = l1w[r * 64 + c];
    else if (c < 128) v = l1b[r * 64 + (c - 64)];
    else              v = root[r * 64 + (c - 128)];
    W[i] = v;
}

// deg[dst] += 1  (hardware f32 atomic)
__global__ void count_deg(const int* __restrict__ ei, float* __restrict__ deg) {
    int e = blockIdx.x * blockDim.x + threadIdx.x;
    if (e >= N_EDGES) return;
    unsafeAtomicAdd(&deg[ei[N_EDGES + e]], 1.0f);
}

// ------------------------------------------------------ WMMA f32 GEMM (fused lrelu)
// out[N,192] = lrelu(h[N,64]) @ W[64,192]
// One wave computes a 16-row x 64-col slab: 4 independent 16x16 accumulator
// chains (ILP across WMMAs), A fragments loaded + activated ONCE and reused
// for all 4 col-tiles. 64 x V_WMMA_F32_16X16X4_F32 per wave.
__global__ __launch_bounds__(32) void gemm_lrelu_wmma(const float* __restrict__ h,
                                                      const float* __restrict__ W,
                                                      float* __restrict__ out) {
    const int lane = threadIdx.x;     // 0..31
    const int g    = lane >> 4;       // lane half: selects K offset (+2) / M offset (+8)
    const int l    = lane & 15;
    const int row0 = blockIdx.x << 4; // N_NODES = 10000 = 625 * 16, exact
    const int col0 = blockIdx.y << 6; // 192 = 3 * 64, exact (4 tiles of 16 per wave)

    const float* arow  = h + (size_t)(row0 + l) * EMBED; // A: row M = row0+l
    const float* bbase = W + col0 + l;                   // B: first col N = col0+l

    // Preload and activate all 16 A fragments (K=64 in steps of 4).
    v2f a[16];
#pragma unroll
    for (int k = 0; k < 16; ++k) {
        const int ka = 4 * k + 2 * g;               // even -> 8-byte-aligned float2
        float2 av = *reinterpret_cast<const float2*>(arow + ka);
        a[k].x = lrelu(av.x);                       // A VGPR0 = K=ka, VGPR1 = K=ka+1
        a[k].y = lrelu(av.y);
    }

    v8f c0 = {0.f,0.f,0.f,0.f,0.f,0.f,0.f,0.f};
    v8f c1 = c0, c2 = c0, c3 = c0;
#pragma unroll
    for (int k = 0; k < 16; ++k) {
        const int ka = 4 * k + 2 * g;
        const float* b0 = bbase + (size_t)ka * WCOLS;   // B row K=ka
        const float* b1 = b0 + WCOLS;                   // B row K=ka+1
        v2f b;
        b.x = b0[0];  b.y = b1[0];
        c0 = __builtin_amdgcn_wmma_f32_16x16x4_f32(false, a[k], false, b, (short)0, c0, false, false);
        b.x = b0[16]; b.y = b1[16];
        c1 = __builtin_amdgcn_wmma_f32_16x16x4_f32(false, a[k], false, b, (short)0, c1, false, false);
        b.x = b0[32]; b.y = b1[32];
        c2 = __builtin_amdgcn_wmma_f32_16x16x4_f32(false, a[k], false, b, (short)0, c2, false, false);
        b.x = b0[48]; b.y = b1[48];
        c3 = __builtin_amdgcn_wmma_f32_16x16x4_f32(false, a[k], false, b, (short)0, c3, false, false);
    }

    // D layout: VGPR v -> row row0 + v + 8g, col col0 + (tile*16) + l
    float* orow = out + (size_t)(row0 + 8 * g) * WCOLS + col0 + l;
#pragma unroll
    for (int v = 0; v < 8; ++v) {
        orow[(size_t)v * WCOLS +  0] = c0[v];
        orow[(size_t)v * WCOLS + 16] = c1[v];
        orow[(size_t)v * WCOLS + 32] = c2[v];
        orow[(size_t)v * WCOLS + 48] = c3[v];
    }
}

// msg[e][o] = (ea[e]/100) * p[src][o] + q[src][o] ; aggr[dst][o] += msg
__global__ void edge_msg(const int* __restrict__ ei, const float* __restrict__ ea,
                         const float* __restrict__ pqr, float* __restrict__ aggr) {
    int i = blockIdx.x * blockDim.x + threadIdx.x;
    if (i >= N_EDGES * EMBED) return;
    int e = i >> 6, o = i & 63;
    int s = ei[e], d = ei[N_EDGES + e];
    float w = ea[e] * 0.01f;
    const float* ps = pqr + (size_t)s * WCOLS;
    float val = fmaf(w, ps[o], ps[64 + o]);
    unsafeAtomicAdd(&aggr[(size_t)d * EMBED + o], val);
}

// h[n][o] = aggr[n][o]/max(deg[n],1) + r[n][o] + bias[o]
__global__ void combine(const float* __restrict__ aggr, const float* __restrict__ deg,
                        const float* __restrict__ pqr, const float* __restrict__ bias,
                        float* __restrict__ h) {
    int i = blockIdx.x * blockDim.x + threadIdx.x;
    if (i >= N_NODES * EMBED) return;
    int n = i >> 6, o = i & 63;
    float dg = fmaxf(deg[n], 1.0f);
    h[i] = aggr[i] / dg + pqr[(size_t)n * WCOLS + 128 + o] + bias[o];
}

// ---------------------------------------------------------------- launcher
extern "C" void kernel_launch(void* const* d_in, const int* in_sizes, int n_in,
                              void* d_out, int out_size, void* d_ws, size_t ws_size,
                              hipStream_t stream) {
    const float* x    = (const float*)d_in[0];
    const int*   ei   = (const int*)  d_in[1]; // [2, E]: row0 = src, row1 = dst
    const float* ea   = (const float*)d_in[2];
    const float* nw   = (const float*)d_in[3];
    const float* nb   = (const float*)d_in[4];
    const float* l1w  = (const float*)d_in[5];
    const float* l1b  = (const float*)d_in[6];
    const float* root = (const float*)d_in[7];
    const float* bias = (const float*)d_in[8];
    float* h = (float*)d_out;                  // h lives in d_out across layers

    float* ws   = (float*)d_ws;
    float* W    = ws;                                    // 64*192      = 12288 f
    float* pqr  = W    + (size_t)EMBED * WCOLS;          // N*192       = 1.92M f
    float* aggr = pqr  + (size_t)N_NODES * WCOLS;        // N*64        = 640K f
    float* deg  = aggr + (size_t)N_NODES * EMBED;        // N           = 10K  f
    // total ~10.3 MB of d_ws

    const int B = 256;
    init_h   <<<(N_NODES * EMBED + B - 1) / B, B, 0, stream>>>(x, nw, nb, h);
    build_w  <<<(EMBED * WCOLS + B - 1) / B,   B, 0, stream>>>(l1w, l1b, root, W);
    zero_f32 <<<(N_NODES + B - 1) / B,         B, 0, stream>>>(deg, N_NODES);
    count_deg<<<(N_EDGES + B - 1) / B,         B, 0, stream>>>(ei, deg);

    dim3 ggrid(N_NODES / 16, WCOLS / 64);      // 625 x 3 wave-slabs (16 x 64 each)
    for (int layer = 0; layer < 3; ++layer) {
        zero_f32       <<<(N_NODES * EMBED + B - 1) / B, B, 0, stream>>>(aggr, N_NODES * EMBED);
        gemm_lrelu_wmma<<<ggrid, dim3(32), 0, stream>>>(h, W, pqr);
        edge_msg       <<<(N_EDGES * EMBED + B - 1) / B, B, 0, stream>>>(ei, ea, pqr, aggr);
        combine        <<<(N_NODES * EMBED + B - 1) / B, B, 0, stream>>>(aggr, deg, pqr, bias, h);
    }
}